// MPSClassifier_987842478103
// MI455X (gfx1250) — compile-verified
//
#include <hip/hip_runtime.h>
#include <math.h>

// ---- problem constants (from reference) ----
#define B_TOT   8192
#define N_SITES 784
#define N_MID   782      // middle sites
#define N_PAD   784      // padded site count (multiple-of-CHUNK staging)
#define BD      10       // bond dim
#define NC      10       // classes
#define PI_F    3.14159265358979323846f

#define SITE_F  384                      // floats per packed site (6 frags * 32 lanes * 2)
#define CHUNK   16                       // sites staged per async chunk
#define NCHUNK  ((N_MID + CHUNK - 1) / CHUNK)        // 49
#define CHUNK_B (CHUNK * SITE_F * 4)                 // 24576 bytes
#define COPIES_PER_THREAD (CHUNK_B / (256 * 16))     // 6 x b128 per thread

typedef __attribute__((ext_vector_type(2))) float v2f;
typedef __attribute__((ext_vector_type(8))) float v8f;

// ---- gfx1250 async global->LDS copy (ASYNCcnt-tracked), inline-asm for
// toolchain portability (bridge doc: builtin arity differs across toolchains).
__device__ __forceinline__ void async_ld16(uint32_t lds_off, const void* g) {
    asm volatile("global_load_async_to_lds_b128 %0, %1, off"
                 :: "v"(lds_off), "v"((unsigned long long)(uintptr_t)g)
                 : "memory");
}

#if __has_builtin(__builtin_amdgcn_s_wait_asynccnt)
#define WAIT_ASYNC(n) __builtin_amdgcn_s_wait_asynccnt(n)
#else
#define WAIT_ASYNC(n) asm volatile("s_wait_asynccnt " #n ::: "memory")
#endif

// ---------------------------------------------------------------------------
// Kernel 0: pack cores_mid (site, l, d, r) into WMMA A-fragments for
// V_WMMA_F32_16X16X4_F32, zero-padded to N_PAD sites.
//   W'[m = 2r+d, l] = cores_mid[site, l, d, r]   (20 x 10, zero-padded 32x12)
// A-fragment element for (tile t, kstep ks, lane ln, vgpr v):
//   row m = 16t + (ln&15), col k = 4ks + v + 2*(ln>>4)
// Stored as frag[((site*6 + t*3+ks)*32 + ln)*2 + v]  -> per-lane b64 loads.
// ---------------------------------------------------------------------------
__global__ void mps_pack_frags(const float* __restrict__ cores_mid,
                               float* __restrict__ frag) {
    int gid = blockIdx.x * blockDim.x + threadIdx.x;
    const int total = N_PAD * 6 * 32;
    if (gid >= total) return;
    int ln   = gid & 31;
    int idx  = (gid >> 5) % 6;          // t*3 + ks
    int site = gid / (6 * 32);
    int t  = idx / 3;
    int m  = 16 * t + (ln & 15);
    int ks = idx % 3;
    float vv[2];
#pragma unroll
    for (int v = 0; v < 2; ++v) {
        int k = 4 * ks + v + 2 * (ln >> 4);
        float val = 0.0f;
        if (site < N_MID && m < 2 * BD && k < BD) {
            int r = m >> 1, d = m & 1;
            val = cores_mid[((site * BD + k) * 2 + d) * BD + r];  // [site][l=k][d][r]
        }
        vv[v] = val;
    }
    float* dst = frag + ((size_t)(site * 6 + idx) * 32 + ln) * 2;
    dst[0] = vv[0];
    dst[1] = vv[1];
}

// ---------------------------------------------------------------------------
// Kernel 1: the MPS chain. One wave32 owns 16 samples; a 256-thread block
// (8 waves, 128 samples) cooperatively double-buffers A-fragments through
// LDS with async global->LDS copies.
// B-fragment semantics (res^T, 12x16, K padded):
//   bf[ks][v] holds component k = 4ks + v + 2*half for sample b0 + (lane&15).
// Per site: 6x v_wmma_f32_16x16x4_f32, then cos/sin recombine + half-swaps.
// ---------------------------------------------------------------------------
__global__ __launch_bounds__(256) void mps_chain(
    const float* __restrict__ x,
    const float* __restrict__ core_first,    // (1,2,10) -> cf[d*10+k]
    const float* __restrict__ frag,          // packed A fragments (N_PAD sites)
    const float* __restrict__ core_last,     // (10,2,1) -> cl[k*2+d]
    const float* __restrict__ output_tensors,// (10,10,1)-> ot[c*10+d]
    float* __restrict__ out)                 // (8192,10)
{
    __shared__ float lbuf[2][CHUNK * SITE_F];   // 2 x 24 KB

    const int tid  = threadIdx.x;
    const int lane = tid & 31;
    const int wave = (blockIdx.x * blockDim.x + tid) >> 5;
    const int bs   = wave * 16 + (lane & 15);   // this lane's sample
    const bool hi  = (lane & 16) != 0;

    const uint32_t lds0 = (uint32_t)(uintptr_t)&lbuf[0][0];
    const uint32_t lds1 = (uint32_t)(uintptr_t)&lbuf[1][0];

    // stage one CHUNK of fragments into an LDS buffer (whole block cooperates)
    auto stage = [&](int chunk, uint32_t lbase) {
        const char* g = (const char*)frag + (size_t)chunk * CHUNK_B + (size_t)tid * 16;
        uint32_t l = lbase + (uint32_t)tid * 16;
#pragma unroll
        for (int i = 0; i < COPIES_PER_THREAD; ++i)
            async_ld16(l + i * 4096, g + i * 4096);
    };

    stage(0, lds0);

    // ---- prologue: res0 = feats[:,0] @ core_first[0], directly in B-frag form
    float xv = x[bs * N_SITES + 0];
    float s0, c0;
    __sincosf(PI_F * xv, &s0, &c0);
    v2f bf[3];
#pragma unroll
    for (int ks = 0; ks < 3; ++ks) {
#pragma unroll
        for (int v = 0; v < 2; ++v) {
            int k = 4 * ks + v + (hi ? 2 : 0);
            float val = 0.0f;
            if (k < BD) val = c0 * core_first[k] + s0 * core_first[BD + k];
            bf[ks][v] = val;
        }
    }

    // ---- main chain over middle sites, CHUNK sites per LDS buffer ----
    int sg = 0;                                // global mid-site index
    for (int c = 0; c < NCHUNK; ++c) {
        if (c + 1 < NCHUNK) {
            stage(c + 1, (c & 1) ? lds0 : lds1);
            WAIT_ASYNC(6);                     // previous chunk's copies done
        } else {
            WAIT_ASYNC(0);
        }
        __syncthreads();                       // publish LDS across all 8 waves

        const float* buf = (c & 1) ? &lbuf[1][0] : &lbuf[0][0];
        const int lim = (N_MID - c * CHUNK < CHUNK) ? (N_MID - c * CHUNK) : CHUNK;
        for (int j = 0; j < lim; ++j, ++sg) {
            const float* fb = buf + (size_t)j * SITE_F + lane * 2;

            // prefetch upcoming x column (global_prefetch_b8)
            __builtin_prefetch(&x[bs * N_SITES + sg + 32], 0, 3);

            v8f acc0 = {}, acc1 = {};
#pragma unroll
            for (int ks = 0; ks < 3; ++ks) {
                v2f a0 = *(const v2f*)(fb + (size_t)(0 * 3 + ks) * 64);  // ds_load_b64
                acc0 = __builtin_amdgcn_wmma_f32_16x16x4_f32(
                    false, a0, false, bf[ks], (short)0, acc0, false, false);
            }
#pragma unroll
            for (int ks = 0; ks < 3; ++ks) {
                v2f a1 = *(const v2f*)(fb + (size_t)(1 * 3 + ks) * 64);
                acc1 = __builtin_amdgcn_wmma_f32_16x16x4_f32(
                    false, a1, false, bf[ks], (short)0, acc1, false, false);
            }

            // feats for this site (x column sg+1)
            xv = x[bs * N_SITES + (sg + 1)];
            float sn, cs;
            __sincosf(PI_F * xv, &sn, &cs);

            // res_new[r] = cos*T'[2r] + sin*T'[2r+1]
            // u_r : half0 -> res_new[r], half1 -> res_new[r+4]
            float u0 = cs * acc0[0] + sn * acc0[1];
            float u1 = cs * acc0[2] + sn * acc0[3];
            float u2 = cs * acc0[4] + sn * acc0[5];
            float u3 = cs * acc0[6] + sn * acc0[7];
            float u4 = cs * acc1[0] + sn * acc1[1];  // h0: res8, h1: 0 (pad)
            float u5 = cs * acc1[2] + sn * acc1[3];  // h0: res9, h1: 0 (pad)

            float w0 = __shfl_xor(u0, 16, 32);
            float w1 = __shfl_xor(u1, 16, 32);
            float w2 = __shfl_xor(u2, 16, 32);
            float w3 = __shfl_xor(u3, 16, 32);

            bf[0][0] = hi ? w2 : u0;   // k0 | k2
            bf[0][1] = hi ? w3 : u1;   // k1 | k3
            bf[1][0] = hi ? u2 : w0;   // k4 | k6
            bf[1][1] = hi ? u3 : w1;   // k5 | k7
            bf[2][0] = u4;             // k8 | pad0
            bf[2][1] = u5;             // k9 | pad0
        }
        __syncthreads();               // all waves done reading this buffer
    }

    // ---- epilogue: s[b] = sum_k res[k] * (cos*cl[k,0] + sin*cl[k,1]) ----
    xv = x[bs * N_SITES + (N_SITES - 1)];
    float sL, cL;
    __sincosf(PI_F * xv, &sL, &cL);
    float partial = 0.0f;
#pragma unroll
    for (int ks = 0; ks < 3; ++ks) {
#pragma unroll
        for (int v = 0; v < 2; ++v) {
            int k = 4 * ks + v + (hi ? 2 : 0);
            if (k < BD)
                partial += bf[ks][v] *
                           (cL * core_last[k * 2 + 0] + sL * core_last[k * 2 + 1]);
        }
    }
    float sb = partial + __shfl_xor(partial, 16, 32);

    // logits[b,c] = s[b] * sum_d output_tensors[c,d,0] (size-1 'd' broadcast)
    if (!hi) {
#pragma unroll
        for (int cc = 0; cc < NC; ++cc) {
            float w = 0.0f;
#pragma unroll
            for (int d = 0; d < BD; ++d) w += output_tensors[cc * BD + d];
            out[bs * NC + cc] = sb * w;
        }
    }
}

// ---------------------------------------------------------------------------
extern "C" void kernel_launch(void* const* d_in, const int* in_sizes, int n_in,
                              void* d_out, int out_size, void* d_ws, size_t ws_size,
                              hipStream_t stream) {
    const float* x          = (const float*)d_in[0];
    const float* core_first = (const float*)d_in[1];
    const float* cores_mid  = (const float*)d_in[2];
    const float* core_last  = (const float*)d_in[3];
    const float* out_t      = (const float*)d_in[4];
    float* out  = (float*)d_out;
    float* frag = (float*)d_ws;   // needs N_PAD*384*4 = 1,204,224 bytes

    {
        int total  = N_PAD * 6 * 32;
        int blocks = (total + 255) / 256;
        mps_pack_frags<<<blocks, 256, 0, stream>>>(cores_mid, frag);
    }
    {
        int waves   = B_TOT / 16;            // 512 waves, 16 samples each
        int threads = waves * 32;            // 16384 -> 64 blocks of 256
        mps_chain<<<threads / 256, 256, 0, stream>>>(
            x, core_first, frag, core_last, out_t, out);
    }
}